// TrajectoryLoss_57793079935554
// MI455X (gfx1250) — compile-verified
//
#include <hip/hip_runtime.h>
#include <hip/hip_bf16.h>
#include <type_traits>

typedef __attribute__((ext_vector_type(2))) float v2f;
typedef __attribute__((ext_vector_type(4))) float v4f;
typedef __attribute__((ext_vector_type(8))) float v8f;

// Problem shape (fixed by the reference harness)
constexpr int B = 2, T = 8, M = 1024, NB = 2048, NRES = 128;
constexpr int NT = M / 16;                 // 64 tiles per axis
constexpr int NTT = NT * (NT + 1) / 2;     // upper-triangular tile count per batch
constexpr float SIGMA_DATA = 16.0f;

// ws layout (floats)
constexpr int OFF_BOND_NUM = 0;
constexpr int OFF_BOND_DEN = 1;
constexpr int OFF_LDDT_NUM = 2;
constexpr int OFF_LDDT_DEN = 3;
constexpr int OFF_NVALID   = 4;                  // B entries
constexpr int OFF_PF       = 8;                  // B*T entries
constexpr int OFF_RES_SUM  = 32;                 // B*NRES
constexpr int OFF_RES_CNT  = 32 + B * NRES;      // B*NRES
constexpr int OFF_NORM_P   = 32 + 2 * B * NRES;  // B*T*M  (16B aligned: 544*4)
constexpr int OFF_NORM_G   = OFF_NORM_P + B * T * M;
constexpr int ZERO_FLOATS  = OFF_NORM_P;         // zero accumulator region only

// ---------------- Kernel 1: row norms + structure-loss partials ----------------
__global__ void __launch_bounds__(256)
k_norm_struct(const float* __restrict__ xp, const float* __restrict__ xg,
              const unsigned char* __restrict__ am, const unsigned char* __restrict__ om,
              const int* __restrict__ mol, float* __restrict__ ws) {
  const int bt = blockIdx.x;            // 0 .. B*T-1
  const int b  = bt / T;
  const int t  = bt % T;
  const int tid = threadIdx.x;
  float accw = 0.f, accv = 0.f;
#pragma unroll
  for (int q = 0; q < M / 256; ++q) {
    int m = tid + q * 256;
    const float* p = xp + ((size_t)bt * M + m) * 3;
    const float* g = xg + ((size_t)bt * M + m) * 3;
    float px = p[0], py = p[1], pz = p[2];
    float gx = g[0], gy = g[1], gz = g[2];
    ws[OFF_NORM_P + bt * M + m] = px * px + py * py + pz * pz;
    ws[OFF_NORM_G + bt * M + m] = gx * gx + gy * gy + gz * gz;
    float dx = px - gx, dy = py - gy, dz = pz - gz;
    float v = (am[b * M + m] && om[b * M + m]) ? 1.f : 0.f;
    int mt = mol[b * M + m];
    float mw = (mt == 3) ? 10.f : ((mt >= 1) ? 5.f : 1.f);   // MOL_W = {1,5,5,10}
    accw += (dx * dx + dy * dy + dz * dz) * mw * v;
    accv += v;
  }
  __shared__ float s1[256], s2[256];
  s1[tid] = accw; s2[tid] = accv;
  __syncthreads();
  for (int s = 128; s > 0; s >>= 1) {
    if (tid < s) { s1[tid] += s1[tid + s]; s2[tid] += s2[tid + s]; }
    __syncthreads();
  }
  if (tid == 0) {
    ws[OFF_PF + bt] = s1[0];
    if (t == 0) ws[OFF_NVALID + b] = s2[0];
  }
}

// ---------------- Kernel 2: bond loss ----------------
__global__ void __launch_bounds__(256)
k_bond(const float* __restrict__ xp, const float* __restrict__ bl,
       const unsigned char* __restrict__ cm, const int* __restrict__ bidx,
       const unsigned char* __restrict__ bm, float* __restrict__ ws) {
  const int id = blockIdx.x * 256 + threadIdx.x;   // < B*T*NB (exact multiple of 256)
  int b = id / (T * NB);
  int t = (id / NB) % T;
  int n = id % NB;
  float w = (!cm[b * T + t] && bm[b * NB + n]) ? 1.f : 0.f;
  int i = bidx[(b * NB + n) * 2 + 0];
  int j = bidx[(b * NB + n) * 2 + 1];
  const size_t base = (size_t)(b * T + t) * M;
  const float* pi = xp + (base + i) * 3;
  const float* pj = xp + (base + j) * 3;
  float dx = pi[0] - pj[0], dy = pi[1] - pj[1], dz = pi[2] - pj[2];
  float len = __fsqrt_rn(dx * dx + dy * dy + dz * dz + 1e-12f);
  float e = len - bl[b * NB + n];
  float num = e * e * w;
  float den = w;
  __shared__ float s1[256], s2[256];
  const int tid = threadIdx.x;
  s1[tid] = num; s2[tid] = den;
  __syncthreads();
  for (int s = 128; s > 0; s >>= 1) {
    if (tid < s) { s1[tid] += s1[tid + s]; s2[tid] += s2[tid + s]; }
    __syncthreads();
  }
  if (tid == 0) {
    atomicAdd(&ws[OFF_BOND_NUM], s1[0]);
    atomicAdd(&ws[OFF_BOND_DEN], s2[0]);
  }
}

// ---------------- Kernel 3: WMMA pairwise tile kernel (lDDT + flex) ----------------
// One wave32 owns one 16x16 (i,j) tile with J >= I (symmetry: off-diagonal lDDT x2).
// D = A(16x4) x B(4x16) via v_wmma_f32_16x16x4_f32, K=3 padded to 4.
__global__ void __launch_bounds__(256)
k_pair(const float* __restrict__ xp, const float* __restrict__ xg,
       const unsigned char* __restrict__ cm, const unsigned char* __restrict__ am,
       const unsigned char* __restrict__ om, const int* __restrict__ resi,
       float* __restrict__ ws) {
  const int lane = threadIdx.x & 31;
  const int wave = threadIdx.x >> 5;
  const int g = blockIdx.x * 8 + wave;           // wave-tile id, < B*NTT
  const int b = g / NTT;
  const int k = g % NTT;
  // decode upper-triangular (I,J), I <= J
  int I = (int)((2.f * NT + 1.f -
                 __fsqrt_rn((float)((2 * NT + 1) * (2 * NT + 1) - 8 * k))) * 0.5f);
  while (I > 0 && I * NT - I * (I - 1) / 2 > k) --I;
  while ((I + 1) * NT - (I + 1) * I / 2 <= k) ++I;
  const int J = I + (k - (I * NT - I * (I - 1) / 2));
  const float fac = (I == J) ? 1.f : 2.f;        // symmetric lDDT doubling

  const int hi = lane >> 4;                      // 0: K=0/1 half, 1: K=2/pad half
  const int lo = lane & 15;
  const float hsel = hi ? 1.f : 0.f;             // arithmetic lane select helper
  const int rowA  = I * 16 + lo;                 // A-fragment row for this lane
  const int rowB  = J * 16 + lo;                 // B-fragment column for this lane
  const int iBase = I * 16 + 8 * hi;             // C rows handled: iBase..iBase+7
  const int j     = J * 16 + lo;                 // C column

  // frame weights from a single 8-byte cond_mask load; no t-indexed arrays
  const unsigned long long cmbits = *(const unsigned long long*)(cm + b * T);
  float cnt = 0.f;
#pragma unroll
  for (int t = 0; t < T; ++t) cnt += ((cmbits >> (8 * t)) & 1ull) ? 0.f : 1.f;
  const float inv = __frcp_rn(fmaxf(cnt, 1.f));  // fw_t = tf_t * inv
  const float S   = cnt * inv;                   // sum of fw over frames

  const float vj = (am[b * M + j] && om[b * M + j]) ? 1.f : 0.f;
  float vi[8]; int ri[8];
#pragma unroll
  for (int r = 0; r < 8; ++r) {
    int i = iBase + r;
    vi[r] = (am[b * M + i] && om[b * M + i]) ? 1.f : 0.f;
    ri[r] = resi[b * M + i];
  }
  const int rj = resi[b * M + j];

  // t-invariant lDDT mask factor and flex pair mask
  float mbase[8], pmv[8];
  int pmAnyLane = 0;
#pragma unroll
  for (int r = 0; r < 8; ++r) {
    int i = iBase + r;
    mbase[r] = vi[r] * vj * (i != j ? 1.f : 0.f) * fac;
    pmv[r]   = (ri[r] == rj ? 1.f : 0.f) * vi[r] * vj * (i < j ? 1.f : 0.f);
    pmAnyLane |= (pmv[r] > 0.f) ? 1 : 0;
  }
  const bool flexAny = (__any(pmAnyLane) != 0);  // wave-uniform

  float sp[8], sp2[8], sg[8], sg2[8];
#pragma unroll
  for (int r = 0; r < 8; ++r) { sp[r] = sp2[r] = sg[r] = sg2[r] = 0.f; }
  float ssum = 0.f, scnt = 0.f;

  // sigmoid(th - ad) = 1 / (1 + e^{ad} * e^{-th})
  const float c0 = 0.60653066f;   // exp(-0.5)
  const float c1 = 0.36787944f;   // exp(-1)
  const float c2 = 0.13533528f;   // exp(-2)
  const float c3 = 0.018315639f;  // exp(-4)

  auto run = [&](auto FLEXC) {
    constexpr bool FLEX = decltype(FLEXC)::value;
#pragma unroll 1
    for (int t = 0; t < T; ++t) {
      const size_t base = (size_t)(b * T + t) * M;

      // unconditional coord loads; lane-half selection via cndmask (no EXEC branches)
      const float* pa = xp + (base + rowA) * 3;
      const float* pb = xp + (base + rowB) * 3;
      const float* ga = xg + (base + rowA) * 3;
      const float* gb = xg + (base + rowB) * 3;
      float pax = pa[0], pay = pa[1], paz = pa[2];
      float pbx = pb[0], pby = pb[1], pbz = pb[2];
      float gax = ga[0], gay = ga[1], gaz = ga[2];
      float gbx = gb[0], gby = gb[1], gbz = gb[2];

      v2f a, bf, a2, b2;
      a[0]  = hi ? paz : pax;   a[1]  = pay - hsel * pay;   // hi -> 0
      bf[0] = hi ? pbz : pbx;   bf[1] = pby - hsel * pby;
      a2[0] = hi ? gaz : gax;   a2[1] = gay - hsel * gay;
      b2[0] = hi ? gbz : gbx;   b2[1] = gby - hsel * gby;

      v8f c = {};
      c = __builtin_amdgcn_wmma_f32_16x16x4_f32(false, a, false, bf, (short)0, c,
                                                false, false);
      v8f cg = {};
      cg = __builtin_amdgcn_wmma_f32_16x16x4_f32(false, a2, false, b2, (short)0, cg,
                                                 false, false);

      // row norms (precomputed, 16B-aligned contiguous loads)
      const float* npRow = ws + OFF_NORM_P + base;
      const float* ngRow = ws + OFF_NORM_G + base;
      const float npj = npRow[j], ngj = ngRow[j];
      v4f npA = *(const v4f*)(npRow + iBase);
      v4f npB = *(const v4f*)(npRow + iBase + 4);
      v4f ngA = *(const v4f*)(ngRow + iBase);
      v4f ngB = *(const v4f*)(ngRow + iBase + 4);

      const float mt = ((cmbits >> (8 * t)) & 1ull) ? 0.f : 1.f;
      const float f  = mt * inv;
#pragma unroll
      for (int r = 0; r < 8; ++r) {
        float npi = (r < 4) ? npA[r & 3] : npB[r & 3];
        float ngi = (r < 4) ? ngA[r & 3] : ngB[r & 3];
        float dp = __fsqrt_rn(fmaxf(npi + npj - 2.f * c[r],  1e-12f));
        float dg = __fsqrt_rn(fmaxf(ngi + ngj - 2.f * cg[r], 1e-12f));
        // smooth lDDT (mask folded arithmetically -> uniform EXEC)
        float mask = (dg < 15.f ? 1.f : 0.f) * mt * mbase[r];
        float ad = fabsf(dp - dg);
        float e  = __expf(ad);
        float sc = 0.25f * (__frcp_rn(1.f + e * c0) + __frcp_rn(1.f + e * c1) +
                            __frcp_rn(1.f + e * c2) + __frcp_rn(1.f + e * c3));
        ssum += sc * mask;
        scnt += mask;
        if constexpr (FLEX) {
          sp[r]  += f * dp;  sp2[r] += f * dp * dp;
          sg[r]  += f * dg;  sg2[r] += f * dg * dg;
        }
      }
    }
  };
  if (flexAny) run(std::integral_constant<bool, true>{});
  else         run(std::integral_constant<bool, false>{});

  // flex finalize: var = E[d^2] + mean^2*(S-2)  (exact for weighted form)
  if (flexAny) {
#pragma unroll
    for (int r = 0; r < 8; ++r) {
      if (pmv[r] > 0.f) {
        float mp = sp[r], mg = sg[r];
        float stdp = __fsqrt_rn(sp2[r] + mp * mp * (S - 2.f) + 1e-8f);
        float stdg = __fsqrt_rn(sg2[r] + mg * mg * (S - 2.f) + 1e-8f);
        float d = stdp - stdg;
        atomicAdd(&ws[OFF_RES_SUM + b * NRES + rj], d * d);
        atomicAdd(&ws[OFF_RES_CNT + b * NRES + rj], 1.f);
      }
    }
  }

  // lDDT wave32 reduction, then one atomic per wave
#pragma unroll
  for (int off = 16; off > 0; off >>= 1) {
    ssum += __shfl_xor(ssum, off, 32);
    scnt += __shfl_xor(scnt, off, 32);
  }
  if (lane == 0) {
    atomicAdd(&ws[OFF_LDDT_NUM], ssum);
    atomicAdd(&ws[OFF_LDDT_DEN], scnt);
  }
}

// ---------------- Kernel 4: combine everything into the scalar loss ----------------
__global__ void k_final(const float* __restrict__ sigma, const unsigned char* __restrict__ cm,
                        const float* __restrict__ ws, float* __restrict__ out) {
  if (threadIdx.x != 0 || blockIdx.x != 0) return;
  // structure loss
  float num = 0.f, den = 0.f;
  for (int b = 0; b < B; ++b) {
    float nv = fmaxf(ws[OFF_NVALID + b], 1.f);
    for (int t = 0; t < T; ++t) {
      float tf = cm[b * T + t] ? 0.f : 1.f;
      float s = sigma[b * T + t];
      float d = s * SIGMA_DATA + 1e-8f;
      float w = (s * s + SIGMA_DATA * SIGMA_DATA) / (d * d) * tf;
      num += w * (ws[OFF_PF + b * T + t] / nv);
      den += tf;
    }
  }
  float l_struct = num / fmaxf(den, 1.f);
  float l_bond = ws[OFF_BOND_NUM] / fmaxf(ws[OFF_BOND_DEN], 1.f);
  float l_lddt = 1.f - ws[OFF_LDDT_NUM] / fmaxf(ws[OFF_LDDT_DEN], 1.f);
  float acc = 0.f, nok = 0.f;
  for (int s = 0; s < B * NRES; ++s) {
    float c = ws[OFF_RES_CNT + s];
    if (c > 0.f) { acc += ws[OFF_RES_SUM + s] / c; nok += 1.f; }
  }
  float l_local = acc / fmaxf(nok, 1.f);
  out[0] = l_struct + l_bond + l_lddt + 4.f * l_local;   // ALPHA_BOND=1, BETA_FLEX*BETA_REL_L=4
}

extern "C" void kernel_launch(void* const* d_in, const int* in_sizes, int n_in,
                              void* d_out, int out_size, void* d_ws, size_t ws_size,
                              hipStream_t stream) {
  const float* xp          = (const float*)d_in[0];
  const float* xg          = (const float*)d_in[1];
  const float* sigma       = (const float*)d_in[2];
  const float* bl          = (const float*)d_in[3];
  const unsigned char* cm  = (const unsigned char*)d_in[4];
  const unsigned char* am  = (const unsigned char*)d_in[5];
  const unsigned char* om  = (const unsigned char*)d_in[6];
  const int* mol           = (const int*)d_in[7];
  const int* resi          = (const int*)d_in[8];
  const int* bidx          = (const int*)d_in[9];
  const unsigned char* bm  = (const unsigned char*)d_in[10];
  float* ws = (float*)d_ws;

  hipMemsetAsync(d_ws, 0, ZERO_FLOATS * sizeof(float), stream);
  k_norm_struct<<<B * T, 256, 0, stream>>>(xp, xg, am, om, mol, ws);
  k_bond<<<(B * T * NB) / 256, 256, 0, stream>>>(xp, bl, cm, bidx, bm, ws);
  k_pair<<<(B * NTT + 7) / 8, 256, 0, stream>>>(xp, xg, cm, am, om, resi, ws);
  k_final<<<1, 1, 0, stream>>>(sigma, cm, ws, (float*)d_out);
}